// CoarseMatching_12601434047081
// MI455X (gfx1250) — compile-verified
//
#include <hip/hip_runtime.h>
#include <cstdint>
#include <cstddef>

// Problem constants
#define NB   2
#define LD   4800      // L == S
#define CD   256       // channels
#define H0   60
#define W0   80
#define BRM  2
#define THR  0.2f
// sim = dot / (C * T) = dot / 25.6
#define SIM_SCALE (1.0f / 25.6f)

#define KC    32       // K panel staged in LDS
#define LSTR  36       // padded LDS row stride (floats) to avoid bank conflicts

typedef float v8f __attribute__((ext_vector_type(8)));
typedef float v2f __attribute__((ext_vector_type(2)));

// ---------------------------------------------------------------------------
// Async staging helpers (CDNA5 GLOBAL_LOAD_ASYNC_TO_LDS, tracked by ASYNCcnt)
// ---------------------------------------------------------------------------
template <int N>
__device__ __forceinline__ void wait_asynccnt() {
#if __has_builtin(__builtin_amdgcn_s_wait_asynccnt)
  __builtin_amdgcn_s_wait_asynccnt(N);
#else
  if constexpr (N == 0) asm volatile("s_wait_asynccnt 0x0" ::: "memory");
  else                  asm volatile("s_wait_asynccnt 0x8" ::: "memory");
#endif
}

// Stage a 64 x KC f32 panel (row stride CD in global) into LDS (row stride LSTR).
// 512 b128 transfers; 128 threads -> 4 async loads per thread (8 instr/wave).
__device__ __forceinline__ void async_stage(const float* __restrict__ gsrc,
                                            float* lds_dst, int tid) {
  #pragma unroll
  for (int it = 0; it < 4; ++it) {
    const int e   = tid + it * 128;   // b128 chunk index 0..511
    const int row = e >> 3;           // 8 chunks per row (KC = 32 floats)
    const int c4  = e & 7;
    const uint32_t ldsa = (uint32_t)(size_t)(lds_dst + row * LSTR + c4 * 4);
    const uint64_t ga   = (uint64_t)(size_t)(gsrc + (size_t)row * CD + c4 * 4);
    asm volatile("global_load_async_to_lds_b128 %0, %1, off"
                 :: "v"(ldsa), "v"(ga) : "memory");
    // warm L2 for the panel after the one being staged
    __builtin_prefetch(gsrc + (size_t)row * CD + c4 * 4 + KC, 0, 0);
  }
}

// ---------------------------------------------------------------------------
// K1: sim = (A . B^T) * SIM_SCALE   via V_WMMA_F32_16X16X4_F32
// block = 128 threads (4 waves), tile = 64x64, grid = (75, 75, 2)
// Double-buffered async-to-LDS panel pipeline.
// ---------------------------------------------------------------------------
__global__ __launch_bounds__(128) void k_gemm_sim(const float* __restrict__ A,
                                                  const float* __restrict__ B,
                                                  float* __restrict__ sim) {
  const int n  = blockIdx.z;
  const int m0 = blockIdx.y * 64;
  const int s0 = blockIdx.x * 64;

  const float* a   = A   + (size_t)n * LD * CD + (size_t)m0 * CD;
  const float* b   = B   + (size_t)n * LD * CD + (size_t)s0 * CD;
  float*       out = sim + (size_t)n * LD * LD;

  __shared__ float lA[2][64 * LSTR];
  __shared__ float lB[2][64 * LSTR];

  const int tid   = threadIdx.x;       // 0..127
  const int wave  = tid >> 5;          // 0..3 -> 16-row strip of the tile
  const int lane  = tid & 31;
  const int l15   = lane & 15;
  const int lhalf = lane & 16;         // 0 or 16

  v8f acc[4] = {v8f{}, v8f{}, v8f{}, v8f{}};

  // prologue: stage panel 0 into buffer 0
  async_stage(a, &lA[0][0], tid);
  async_stage(b, &lB[0][0], tid);

  for (int kb = 0; kb < CD; kb += KC) {
    const int cur = (kb / KC) & 1;
    __syncthreads();   // all waves done reading the buffer we overwrite next
    if (kb + KC < CD) {
      async_stage(a + kb + KC, &lA[cur ^ 1][0], tid);
      async_stage(b + kb + KC, &lB[cur ^ 1][0], tid);
      wait_asynccnt<8>();   // async loads complete in order: current panel done
    } else {
      wait_asynccnt<0>();
    }
    __syncthreads();   // cross-wave visibility of the current panel

    const float* pA = &lA[cur][0];
    const float* pB = &lB[cur][0];
    #pragma unroll
    for (int k = 0; k < KC; k += 4) {
      // A frag (16x4 f32): lanes 0-15 hold K=k,k+1; lanes 16-31 hold K=k+2,k+3
      const int akoff = k + (lhalf ? 2 : 0);
      v2f af = *(const v2f*)(pA + (wave * 16 + l15) * LSTR + akoff);
      #pragma unroll
      for (int j = 0; j < 4; ++j) {
        // B frag (4x16 f32): lane n holds column n = feat1 row (B^T)
        v2f bf = *(const v2f*)(pB + (j * 16 + l15) * LSTR + akoff);
        acc[j] = __builtin_amdgcn_wmma_f32_16x16x4_f32(
            false, af, false, bf, (short)0, acc[j], false, false);
      }
    }
  }

  // C/D layout: VGPR v, lanes 0-15 -> M=v, N=lane; lanes 16-31 -> M=8+v
  const int row_base = m0 + wave * 16 + (lhalf ? 8 : 0);
  #pragma unroll
  for (int j = 0; j < 4; ++j) {
    const int col = s0 + j * 16 + l15;
    #pragma unroll
    for (int v = 0; v < 8; ++v) {
      out[(size_t)(row_base + v) * LD + col] = acc[j][v] * SIM_SCALE;
    }
  }
}

// ---------------------------------------------------------------------------
// online softmax helpers
// ---------------------------------------------------------------------------
__device__ __forceinline__ void online_acc(float& m, float& s, float x) {
  if (x > m) { s = s * __expf(m - x) + 1.0f; m = x; }
  else       { s += __expf(x - m); }
}
__device__ __forceinline__ void online_merge(float& m, float& s, float m2, float s2) {
  float mm = fmaxf(m, m2);
  s = s * __expf(m - mm) + s2 * __expf(m2 - mm);
  m = mm;
}

// K2: per-row (axis=2) max/sum(exp);  grid = (4800, 2), block = 256
__global__ __launch_bounds__(256) void k_row_stats(const float* __restrict__ sim,
                                                   float* __restrict__ rmax,
                                                   float* __restrict__ rsum,
                                                   unsigned* __restrict__ rowcmax) {
  const int n = blockIdx.y, i = blockIdx.x;
  const float* row = sim + ((size_t)n * LD + i) * LD;
  float m = -3.4e38f, s = 0.0f;
  for (int j = threadIdx.x; j < LD; j += 256) online_acc(m, s, row[j]);

  __shared__ float sm[256], ss[256];
  sm[threadIdx.x] = m; ss[threadIdx.x] = s;
  __syncthreads();
  for (int off = 128; off > 0; off >>= 1) {
    if ((int)threadIdx.x < off) {
      float m1 = sm[threadIdx.x], s1 = ss[threadIdx.x];
      online_merge(m1, s1, sm[threadIdx.x + off], ss[threadIdx.x + off]);
      sm[threadIdx.x] = m1; ss[threadIdx.x] = s1;
    }
    __syncthreads();
  }
  if (threadIdx.x == 0) {
    rmax[n * LD + i] = sm[0];
    rsum[n * LD + i] = ss[0];
    rowcmax[n * LD + i] = 0u;   // init for later conf atomic-max
  }
}

// K3: column partial stats over a 300-row chunk; grid = (19, 16, 2), block 256
__global__ __launch_bounds__(256) void k_col_partial(const float* __restrict__ sim,
                                                     float* __restrict__ pmax,
                                                     float* __restrict__ psum) {
  const int n = blockIdx.z, chunk = blockIdx.y;
  const int j = blockIdx.x * 256 + threadIdx.x;
  if (j >= LD) return;
  const float* base = sim + (size_t)n * LD * LD;
  const int i0 = chunk * 300;
  float m = -3.4e38f, s = 0.0f;
  for (int i = i0; i < i0 + 300; ++i) online_acc(m, s, base[(size_t)i * LD + j]);
  pmax[((size_t)n * 16 + chunk) * LD + j] = m;
  psum[((size_t)n * 16 + chunk) * LD + j] = s;
}

// K4: combine 16 column partials; grid = (19, 2), block 256
__global__ __launch_bounds__(256) void k_col_combine(const float* __restrict__ pmax,
                                                     const float* __restrict__ psum,
                                                     float* __restrict__ cmax,
                                                     float* __restrict__ csum,
                                                     unsigned* __restrict__ colcmax) {
  const int n = blockIdx.y;
  const int j = blockIdx.x * 256 + threadIdx.x;
  if (j >= LD) return;
  float m = -3.4e38f, s = 0.0f;
  for (int c = 0; c < 16; ++c)
    online_merge(m, s, pmax[((size_t)n * 16 + c) * LD + j],
                       psum[((size_t)n * 16 + c) * LD + j]);
  cmax[n * LD + j] = m;
  csum[n * LD + j] = s;
  colcmax[n * LD + j] = 0u;
}

// K5: conf = exp(2s - rm_i - cm_j)/(rs_i*cs_j) in place; track row/col conf max
// grid = (75, 75, 2), block = 256, tile 64x64
__global__ __launch_bounds__(256) void k_conf(float* __restrict__ conf,
                                              const float* __restrict__ rmax,
                                              const float* __restrict__ rsum,
                                              const float* __restrict__ cmax,
                                              const float* __restrict__ csum,
                                              unsigned* __restrict__ rowcmax,
                                              unsigned* __restrict__ colcmax) {
  const int n = blockIdx.z;
  const int i0 = blockIdx.y * 64, j0 = blockIdx.x * 64;
  __shared__ unsigned trm[64], tcm[64];
  __shared__ float srm[64], srs[64], scm[64], scs[64];
  if (threadIdx.x < 64) {
    trm[threadIdx.x] = 0u; tcm[threadIdx.x] = 0u;
    srm[threadIdx.x] = rmax[n * LD + i0 + threadIdx.x];
    srs[threadIdx.x] = rsum[n * LD + i0 + threadIdx.x];
    scm[threadIdx.x] = cmax[n * LD + j0 + threadIdx.x];
    scs[threadIdx.x] = csum[n * LD + j0 + threadIdx.x];
  }
  __syncthreads();
  float* base = conf + (size_t)n * LD * LD;
  for (int e = threadIdx.x; e < 64 * 64; e += 256) {
    const int r = e >> 6, c = e & 63;
    const size_t idx = (size_t)(i0 + r) * LD + (j0 + c);
    const float s = base[idx];
    const float v = __expf(2.0f * s - srm[r] - scm[c]) / (srs[r] * scs[c]);
    base[idx] = v;
    const unsigned uv = __float_as_uint(v);   // v >= 0: bits order-preserving
    atomicMax(&trm[r], uv);
    atomicMax(&tcm[c], uv);
  }
  __syncthreads();
  if (threadIdx.x < 64) {
    atomicMax(&rowcmax[n * LD + i0 + threadIdx.x], trm[threadIdx.x]);
    atomicMax(&colcmax[n * LD + j0 + threadIdx.x], tcm[threadIdx.x]);
  }
}

// K6: threshold + border + mutual-NN; first surviving j per row.
// grid = (4800, 2), block = 256
__global__ __launch_bounds__(256) void k_match(const float* __restrict__ conf,
                                               const float* __restrict__ rowcmax,
                                               const float* __restrict__ colcmax,
                                               float* __restrict__ out_maskv,
                                               float* __restrict__ out_jids,
                                               float* __restrict__ out_mconf) {
  const int n = blockIdx.y, i = blockIdx.x;
  const int hi = i / W0, wi = i % W0;
  const bool brow = (hi >= BRM) & (hi < H0 - BRM) & (wi >= BRM) & (wi < W0 - BRM);
  const float* row = conf + ((size_t)n * LD + i) * LD;
  const float rmx = rowcmax[n * LD + i];

  int best = 0x7fffffff;
  if (brow) {
    for (int j = threadIdx.x; j < LD; j += 256) {
      const int hj = j / W0, wj = j % W0;
      const bool bcol = (hj >= BRM) & (hj < H0 - BRM) & (wj >= BRM) & (wj < W0 - BRM);
      const float v = row[j];
      if (bcol && (v > THR) && (v == rmx) && (v == colcmax[n * LD + j]))
        best = min(best, j);
    }
  }
  __shared__ int sb[256];
  sb[threadIdx.x] = best;
  __syncthreads();
  for (int off = 128; off > 0; off >>= 1) {
    if ((int)threadIdx.x < off) sb[threadIdx.x] = min(sb[threadIdx.x], sb[threadIdx.x + off]);
    __syncthreads();
  }
  if (threadIdx.x == 0) {
    const int j = sb[0];
    const bool hit = (j != 0x7fffffff);
    out_maskv[n * LD + i] = hit ? 1.0f : 0.0f;
    out_jids [n * LD + i] = hit ? (float)j : 0.0f;
    out_mconf[n * LD + i] = hit ? row[j] : 0.0f;
  }
}

// ---------------------------------------------------------------------------
extern "C" void kernel_launch(void* const* d_in, const int* in_sizes, int n_in,
                              void* d_out, int out_size, void* d_ws, size_t ws_size,
                              hipStream_t stream) {
  const float* f0 = (const float*)d_in[0];   // [2,4800,256]
  const float* f1 = (const float*)d_in[1];   // [2,4800,256]

  float* conf  = (float*)d_out;                       // [2,4800,4800]
  float* maskv = conf + (size_t)NB * LD * LD;         // [2,4800]
  float* jids  = maskv + NB * LD;                     // [2,4800]
  float* mconf = jids + NB * LD;                      // [2,4800]

  float* ws      = (float*)d_ws;
  float* rmax    = ws;               // [2*4800]
  float* rsum    = ws + 1 * NB * LD;
  float* cmax    = ws + 2 * NB * LD;
  float* csum    = ws + 3 * NB * LD;
  float* rowcmax = ws + 4 * NB * LD; // bit-pattern max (uint view)
  float* colcmax = ws + 5 * NB * LD;
  float* pmax    = ws + 6 * NB * LD; // [2*16*4800]
  float* psum    = pmax + (size_t)NB * 16 * LD;

  k_gemm_sim<<<dim3(75, 75, NB), 128, 0, stream>>>(f0, f1, conf);
  k_row_stats<<<dim3(LD, NB), 256, 0, stream>>>(conf, rmax, rsum, (unsigned*)rowcmax);
  k_col_partial<<<dim3(19, 16, NB), 256, 0, stream>>>(conf, pmax, psum);
  k_col_combine<<<dim3(19, NB), 256, 0, stream>>>(pmax, psum, cmax, csum, (unsigned*)colcmax);
  k_conf<<<dim3(75, 75, NB), 256, 0, stream>>>(conf, rmax, rsum, cmax, csum,
                                               (unsigned*)rowcmax, (unsigned*)colcmax);
  k_match<<<dim3(LD, NB), 256, 0, stream>>>(conf, rowcmax, colcmax, maskv, jids, mconf);
}